// UGFM_2095944041033
// MI455X (gfx1250) — compile-verified
//
#include <hip/hip_runtime.h>

#define DD 128
#define HH 8
#define KK 16
#define LL 16
#define NBLK 2
#define NN 100000
#define EE 800000
#define MN_ 1000
#define ME_ 8000
#define VP_ 15
#define BT_ 256

#define SC_D 0.08838834764831845f   /* 1/sqrt(128) */
#define SC_K 0.25f                  /* 1/sqrt(16)  */

typedef float v2f __attribute__((ext_vector_type(2)));
typedef float v8f __attribute__((ext_vector_type(8)));

enum { GM_PLAIN = 0, GM_QMOD = 1, GM_RELU = 2, GM_RELU_RES = 3 };

// ---- ordered-uint encoding for float atomic max ----
__device__ __forceinline__ unsigned encF(float f) {
  unsigned u = __float_as_uint(f);
  return u ^ ((u & 0x80000000u) ? 0xFFFFFFFFu : 0x80000000u);
}
__device__ __forceinline__ float decF(unsigned e) {
  unsigned u = e ^ ((e & 0x80000000u) ? 0x80000000u : 0xFFFFFFFFu);
  return __uint_as_float(u);
}

// ---- row gather: out[i,:] = emb[idx[i],:] (float4 granular) ----
__global__ __launch_bounds__(256)
void k_gather(const float* __restrict__ emb, const int* __restrict__ idx,
              float* __restrict__ outp, int n) {
  int i = blockIdx.x * 256 + threadIdx.x;
  if (i >= n * 32) return;
  int r = i >> 5, c = i & 31;
  reinterpret_cast<float4*>(outp)[(size_t)r * 32 + c] =
      reinterpret_cast<const float4*>(emb)[(size_t)idx[r] * 32 + c];
}

// ---- WMMA f32 GEMM: Y[M,128] = X[M,128] @ W[128,128], fused epilogues ----
// 32 rows x 128 cols per block; each wave owns one 16-col tile and TWO row
// tiles (two independent accumulator chains sharing each B fragment).
__global__ __launch_bounds__(256)
void k_gemm128(const float* __restrict__ X, const float* __restrict__ W,
               float* __restrict__ Y, int M, int mode,
               const float* __restrict__ basev, const int* __restrict__ mid,
               const float* __restrict__ padd) {
  __shared__ float Xs[32 * 132];   // padded stride kills LDS bank conflicts
  const int row0 = blockIdx.x << 5;
  const int t = threadIdx.x;
  for (int i = t; i < 1024; i += 256) {         // 1024 float4 = 32x128 tile
    int r = i >> 5, c4 = i & 31;
    float4 val = make_float4(0.f, 0.f, 0.f, 0.f);
    if (row0 + r < M)
      val = reinterpret_cast<const float4*>(X)[(size_t)(row0 + r) * 32 + c4];
    reinterpret_cast<float4*>(Xs + r * 132)[c4] = val;
  }
  __syncthreads();
  const int wave = t >> 5, lane = t & 31;
  const int mrow = lane & 15;
  const int khalf = (lane >> 4) << 1;           // lanes 16-31 hold K+2,K+3
  const int col = (wave << 4) + mrow;           // each wave owns one 16-col tile
  v8f acc0 = {0.f, 0.f, 0.f, 0.f, 0.f, 0.f, 0.f, 0.f};
  v8f acc1 = {0.f, 0.f, 0.f, 0.f, 0.f, 0.f, 0.f, 0.f};
  for (int k0 = 0; k0 < 128; k0 += 4) {
    v2f a0, a1, b;
    b.x = W[(k0 + khalf) * 128 + col];          // B frag shared by both tiles
    b.y = W[(k0 + khalf + 1) * 128 + col];
    a0.x = Xs[mrow * 132 + k0 + khalf];
    a0.y = Xs[mrow * 132 + k0 + khalf + 1];
    a1.x = Xs[(16 + mrow) * 132 + k0 + khalf];
    a1.y = Xs[(16 + mrow) * 132 + k0 + khalf + 1];
    acc0 = __builtin_amdgcn_wmma_f32_16x16x4_f32(false, a0, false, b,
                                                 (short)0, acc0, false, false);
    acc1 = __builtin_amdgcn_wmma_f32_16x16x4_f32(false, a1, false, b,
                                                 (short)0, acc1, false, false);
  }
  const int half8 = (lane >> 4) << 3;           // lanes 16-31 hold rows M+8..M+15
#pragma unroll
  for (int tile = 0; tile < 2; ++tile) {
    const int rbase = row0 + (tile << 4) + half8;
    v8f acc = tile ? acc1 : acc0;
    for (int r = 0; r < 8; ++r) {
      int rr = rbase + r;
      if (rr >= M) break;
      float g = acc[r];
      float* yp = Y + (size_t)rr * 128 + col;
      if (mode == GM_PLAIN) {
        *yp = g;
      } else if (mode == GM_QMOD) {
        float f = basev[col];
        if (padd) f += padd[(size_t)mid[rr] * 128 + col];
        *yp = g * f;
      } else if (mode == GM_RELU) {
        *yp = fmaxf(g, 0.f);
      } else {                                  // GM_RELU_RES: in-place residual
        *yp = *yp + fmaxf(g, 0.f);
      }
    }
  }
}

// ---- meta-graph edge score: ke = kf[src]*base_e2 (stored), score = q[dst].ke/4 ----
__global__ __launch_bounds__(256)
void k_meta_score(const float* __restrict__ q, const float* __restrict__ kf,
                  const float* __restrict__ base_e2,
                  const int* __restrict__ src, const int* __restrict__ dst,
                  float* __restrict__ medge, float* __restrict__ score,
                  unsigned* __restrict__ mEnc, int E) {
  int idx = blockIdx.x * 256 + threadIdx.x;
  if (idx >= E * HH) return;
  int e = idx >> 3, h = idx & 7;
  int s = src[e], d = dst[e];
  const float* kp = kf + (size_t)s * 128 + h * 16;
  const float* qp = q + (size_t)d * 128 + h * 16;
  const float* be = base_e2 + h * 16;
  float* mp = medge + (size_t)e * 128 + h * 16;
  float acc = 0.f;
#pragma unroll
  for (int k = 0; k < 16; ++k) {
    float ke = kp[k] * be[k];
    mp[k] = ke;
    acc += qp[k] * ke;
  }
  float sc = acc * SC_K;   // +base_e1,+p1n[dst] uniform per segment -> softmax-invariant
  score[idx] = sc;
  atomicMax(&mEnc[(size_t)d * 8 + h], encF(sc));
}

// ---- main-graph edge score ----
__global__ __launch_bounds__(256)
void k_main_score(const float* __restrict__ q, const float* __restrict__ kf,
                  const float* __restrict__ base_e2, const float* __restrict__ p2me,
                  const float* __restrict__ base_e1, const float* __restrict__ p1me,
                  const int* __restrict__ src, const int* __restrict__ dst,
                  const int* __restrict__ meid,
                  float* __restrict__ score, unsigned* __restrict__ mEnc, int E) {
  int idx = blockIdx.x * 256 + threadIdx.x;
  if (idx >= E * HH) return;
  int e = idx >> 3, h = idx & 7;
  int s = src[e], d = dst[e], me = meid[e];
  const float* kp = kf + (size_t)s * 128 + h * 16;
  const float* qp = q + (size_t)d * 128 + h * 16;
  const float* pe = p2me + (size_t)me * 128 + h * 16;
  const float* be = base_e2 + h * 16;
  float acc = 0.f;
#pragma unroll
  for (int k = 0; k < 16; ++k)
    acc += qp[k] * (kp[k] * (be[k] + pe[k]));
  float sc = acc * SC_K + base_e1[h] + p1me[(size_t)me * 8 + h];
  score[idx] = sc;
  atomicMax(&mEnc[(size_t)d * 8 + h], encF(sc));
}

// ---- segment softmax pass 2: w = exp(s - max); z[dst] += w ----
__global__ __launch_bounds__(256)
void k_exp(float* __restrict__ score, const unsigned* __restrict__ mEnc,
           float* __restrict__ z, const int* __restrict__ dst, int E) {
  int idx = blockIdx.x * 256 + threadIdx.x;
  if (idx >= E * HH) return;
  int e = idx >> 3, h = idx & 7;
  int d = dst[e];
  float m = decF(mEnc[(size_t)d * 8 + h]);
  float w = expf(score[idx] - m);
  score[idx] = w;
  atomicAdd(&z[(size_t)d * 8 + h], w);
}

// ---- pass 3: agg[dst] += (w/(z+eps)) * v[src] ----
__global__ __launch_bounds__(256)
void k_agg(const float* __restrict__ w, const float* __restrict__ z,
           const float* __restrict__ v, const int* __restrict__ src,
           const int* __restrict__ dst, float* __restrict__ agg, int E) {
  int idx = blockIdx.x * 256 + threadIdx.x;
  if (idx >= E * HH) return;
  int e = idx >> 3, h = idx & 7;
  int s = src[e], d = dst[e];
  float a = w[idx] / (z[(size_t)d * 8 + h] + 1e-9f);
  const float* vp = v + (size_t)s * 128 + h * 16;
  float* ap = agg + (size_t)d * 128 + h * 16;
#pragma unroll
  for (int k = 0; k < 16; ++k) atomicAdd(&ap[k], a * vp[k]);
}

// ---- meta learner: w=softmax(X@u.T/sqrt(D)); p2=w@W2; p1=w@W1 ----
__global__ __launch_bounds__(256)
void k_metalearn(const float* __restrict__ X, const float* __restrict__ u,
                 const float* __restrict__ W2, const float* __restrict__ W1,
                 float* __restrict__ p2, float* __restrict__ p1, int n) {
  __shared__ float Us[LL * 128];
  __shared__ float W2s[LL * 128];
  __shared__ float W1s[LL * 8];
  int t = threadIdx.x;
  for (int i = t; i < LL * 128; i += 256) { Us[i] = u[i]; W2s[i] = W2[i]; }
  for (int i = t; i < LL * 8; i += 256) W1s[i] = W1[i];
  __syncthreads();
  int row = blockIdx.x * 256 + t;
  if (row >= n) return;
  const float* x = X + (size_t)row * 128;
  float wv[LL];
#pragma unroll
  for (int l = 0; l < LL; ++l) wv[l] = 0.f;
  for (int d = 0; d < 128; ++d) {
    float xv = x[d];
#pragma unroll
    for (int l = 0; l < LL; ++l) wv[l] += xv * Us[l * 128 + d];
  }
  float mx = -3.4e38f;
#pragma unroll
  for (int l = 0; l < LL; ++l) { wv[l] *= SC_D; mx = fmaxf(mx, wv[l]); }
  float zs = 0.f;
#pragma unroll
  for (int l = 0; l < LL; ++l) { wv[l] = expf(wv[l] - mx); zs += wv[l]; }
  float inv = 1.f / zs;
#pragma unroll
  for (int l = 0; l < LL; ++l) wv[l] *= inv;
#pragma unroll
  for (int h = 0; h < 8; ++h) {
    float a = 0.f;
#pragma unroll
    for (int l = 0; l < LL; ++l) a += wv[l] * W1s[l * 8 + h];
    p1[(size_t)row * 8 + h] = a;
  }
  for (int d = 0; d < 128; ++d) {
    float a = 0.f;
#pragma unroll
    for (int l = 0; l < LL; ++l) a += wv[l] * W2s[l * 128 + d];
    p2[(size_t)row * 128 + d] = a;
  }
}

// ---- residual add: y += x ----
__global__ __launch_bounds__(256)
void k_add(float* __restrict__ y, const float* __restrict__ x, int n) {
  int i = blockIdx.x * 256 + threadIdx.x;
  if (i < n) y[i] += x[i];
}

// ---- static param-name graph init: p2s[15,128], p1s[15,8] ----
__global__ __launch_bounds__(128)
void k_static(const float* __restrict__ emb, const float* __restrict__ u,
              const float* __restrict__ W2, const float* __restrict__ W1,
              const int* __restrict__ svals, const int* __restrict__ ssrc,
              const int* __restrict__ sdst,
              float* __restrict__ p2s, float* __restrict__ p1s) {
  __shared__ float sagg[VP_ * 128];
  __shared__ float red[LL];
  __shared__ float wv[LL];
  int t = threadIdx.x;  // 128 threads, one per feature dim
  for (int i = t; i < VP_ * 128; i += 128) sagg[i] = 0.f;
  __syncthreads();
  for (int ei = 0; ei < 10; ++ei) {  // copy_u/sum over 10 static edges
    int srow = ssrc[ei], drow = sdst[ei];
    sagg[drow * 128 + t] += emb[(size_t)svals[srow] * 128 + t];
  }
  __syncthreads();
  for (int vtx = 0; vtx < VP_; ++vtx) {
    if (t < LL) {
      float a = 0.f;
      for (int d = 0; d < 128; ++d) a += sagg[vtx * 128 + d] * u[t * 128 + d];
      red[t] = a * SC_D;
    }
    __syncthreads();
    if (t == 0) {
      float mx = -3.4e38f;
      for (int l = 0; l < LL; ++l) mx = fmaxf(mx, red[l]);
      float zs = 0.f;
      for (int l = 0; l < LL; ++l) { wv[l] = expf(red[l] - mx); zs += wv[l]; }
      float inv = 1.f / zs;
      for (int l = 0; l < LL; ++l) wv[l] *= inv;
    }
    __syncthreads();
    {
      float a = 0.f;
      for (int l = 0; l < LL; ++l) a += wv[l] * W2[l * 128 + t];
      p2s[vtx * 128 + t] = a;
    }
    if (t < 8) {
      float a = 0.f;
      for (int l = 0; l < LL; ++l) a += wv[l] * W1[l * 8 + t];
      p1s[vtx * 8 + t] = a;
    }
    __syncthreads();
  }
}

// ---- target readout: out[t] += scale * sum_hk tw[h]*feat[idx,h,k] ----
__global__ __launch_bounds__(256)
void k_logit(const float* __restrict__ feat, const float* __restrict__ p1mn,
             const int* __restrict__ mid, const int* __restrict__ tidx,
             const float* __restrict__ base_tw, float* __restrict__ out,
             float scale) {
  int t = threadIdx.x;
  if (t >= BT_) return;
  int i = tidx[t];
  const float* f = feat + (size_t)i * 128;
  const float* pm = p1mn + (size_t)mid[i] * 8;
  float a = 0.f;
#pragma unroll
  for (int h = 0; h < 8; ++h) {
    float tw = base_tw[h] + pm[h];
#pragma unroll
    for (int k = 0; k < 16; ++k) a += tw * f[h * 16 + k];
  }
  out[t] += a * scale;
}

extern "C" void kernel_launch(void* const* d_in, const int* in_sizes, int n_in,
                              void* d_out, int out_size, void* d_ws, size_t ws_size,
                              hipStream_t stream) {
  (void)in_sizes; (void)n_in; (void)out_size; (void)ws_size;
  const float* emb   = (const float*)d_in[0];
  const float* u     = (const float*)d_in[1];
  const float* W2lat = (const float*)d_in[2];
  const float* W1lat = (const float*)d_in[3];
  const float* Wq_m  = (const float*)d_in[4];
  const float* Wk_m  = (const float*)d_in[5];
  const float* Wv_m  = (const float*)d_in[6];
  const float* Wo_m  = (const float*)d_in[7];
  const float* Wq    = (const float*)d_in[8];
  const float* Wk    = (const float*)d_in[9];
  const float* Wv    = (const float*)d_in[10];
  const float* Wo    = (const float*)d_in[11];
  const int* node_vals      = (const int*)d_in[12];
  const int* meta_node_vals = (const int*)d_in[13];
  const int* src  = (const int*)d_in[14];
  const int* dst  = (const int*)d_in[15];
  const int* msrc = (const int*)d_in[16];
  const int* mdst = (const int*)d_in[17];
  const int* mid  = (const int*)d_in[18];
  const int* meid = (const int*)d_in[19];
  const int* tidx = (const int*)d_in[20];
  const int* svals = (const int*)d_in[21];
  const int* ssrc  = (const int*)d_in[22];
  const int* sdst  = (const int*)d_in[23];
  float* out = (float*)d_out;

  size_t off = 0;
  float* base = (float*)d_ws;
  auto carve = [&](size_t n) -> float* { float* r = base + off; off += n; return r; };
  float* feat  = carve((size_t)NN * 128);
  float* kf    = carve((size_t)NN * 128);
  float* vbuf  = carve((size_t)NN * 128);
  float* qagg  = carve((size_t)NN * 128);   // q, later reused as agg
  float* score = carve((size_t)EE * 8);
  float* z     = carve((size_t)NN * 8);
  unsigned* mEnc = (unsigned*)carve((size_t)NN * 8);
  float* mfeat = carve((size_t)MN_ * 128);
  float* mconv = carve((size_t)MN_ * 128);
  float* mq    = carve((size_t)MN_ * 128);
  float* mkf   = carve((size_t)MN_ * 128);
  float* mv    = carve((size_t)MN_ * 128);
  float* magg  = carve((size_t)MN_ * 128);
  float* medge = carve((size_t)ME_ * 128);
  float* mscore = carve((size_t)ME_ * 8);
  float* mz    = carve((size_t)MN_ * 8);
  unsigned* mmEnc = (unsigned*)carve((size_t)MN_ * 8);
  float* p2mn  = carve((size_t)MN_ * 128);
  float* p1mn  = carve((size_t)MN_ * 8);
  float* p2me  = carve((size_t)ME_ * 128);
  float* p1me  = carve((size_t)ME_ * 8);
  float* p2s   = carve((size_t)VP_ * 128);
  float* p1s   = carve((size_t)VP_ * 8);

  const float* base_tw = p1s + 0 * 8;
  const float* base_n2 = p2s + 6 * 128;
  const float* base_e1 = p1s + 9 * 8;
  const float* base_e2 = p2s + 12 * 128;

  hipMemsetAsync(d_out, 0, BT_ * sizeof(float), stream);

  k_gather<<<(NN * 32 + 255) / 256, 256, 0, stream>>>(emb, node_vals, feat, NN);
  k_gather<<<(MN_ * 32 + 255) / 256, 256, 0, stream>>>(emb, meta_node_vals, mfeat, MN_);
  k_static<<<1, 128, 0, stream>>>(emb, u, W2lat, W1lat, svals, ssrc, sdst, p2s, p1s);

  const int gMN = (MN_ + 31) / 32;
  const int gN = (NN + 31) / 32;
  const int gMEH = (ME_ * HH + 255) / 256;
  const int gEH = (EE * HH + 255) / 256;

  for (int b = 0; b < NBLK; ++b) {
    const size_t wo_off = (size_t)b * 128 * 128;
    // ---- meta conv ----
    k_gemm128<<<gMN, 256, 0, stream>>>(mfeat, Wq_m + wo_off, mq, MN_, GM_QMOD,
                                       base_n2, nullptr, nullptr);
    k_gemm128<<<gMN, 256, 0, stream>>>(mfeat, Wk_m + wo_off, mkf, MN_, GM_PLAIN,
                                       nullptr, nullptr, nullptr);
    k_gemm128<<<gMN, 256, 0, stream>>>(mfeat, Wv_m + wo_off, mv, MN_, GM_PLAIN,
                                       nullptr, nullptr, nullptr);
    hipMemsetAsync(mmEnc, 0, (size_t)MN_ * 8 * 4, stream);
    hipMemsetAsync(mz, 0, (size_t)MN_ * 8 * 4, stream);
    hipMemsetAsync(magg, 0, (size_t)MN_ * 128 * 4, stream);
    k_meta_score<<<gMEH, 256, 0, stream>>>(mq, mkf, base_e2, msrc, mdst,
                                           medge, mscore, mmEnc, ME_);
    k_exp<<<gMEH, 256, 0, stream>>>(mscore, mmEnc, mz, mdst, ME_);
    k_agg<<<gMEH, 256, 0, stream>>>(mscore, mz, mv, msrc, mdst, magg, ME_);
    k_gemm128<<<gMN, 256, 0, stream>>>(magg, Wo_m + wo_off, mconv, MN_, GM_RELU,
                                       nullptr, nullptr, nullptr);
    // ---- meta learner on conv outputs ----
    k_metalearn<<<(MN_ + 255) / 256, 256, 0, stream>>>(mconv, u, W2lat, W1lat,
                                                       p2mn, p1mn, MN_);
    k_metalearn<<<(ME_ + 255) / 256, 256, 0, stream>>>(medge, u, W2lat, W1lat,
                                                       p2me, p1me, ME_);
    k_add<<<(MN_ * 128 + 255) / 256, 256, 0, stream>>>(mfeat, mconv, MN_ * 128);
    // ---- main conv ----
    k_gemm128<<<gN, 256, 0, stream>>>(feat, Wq + wo_off, qagg, NN, GM_QMOD,
                                      base_n2, mid, p2mn);
    k_gemm128<<<gN, 256, 0, stream>>>(feat, Wk + wo_off, kf, NN, GM_PLAIN,
                                      nullptr, nullptr, nullptr);
    k_gemm128<<<gN, 256, 0, stream>>>(feat, Wv + wo_off, vbuf, NN, GM_PLAIN,
                                      nullptr, nullptr, nullptr);
    hipMemsetAsync(mEnc, 0, (size_t)NN * 8 * 4, stream);
    hipMemsetAsync(z, 0, (size_t)NN * 8 * 4, stream);
    k_main_score<<<gEH, 256, 0, stream>>>(qagg, kf, base_e2, p2me, base_e1, p1me,
                                          src, dst, meid, score, mEnc, EE);
    k_exp<<<gEH, 256, 0, stream>>>(score, mEnc, z, dst, EE);
    hipMemsetAsync(qagg, 0, (size_t)NN * 128 * 4, stream);   // q dead -> agg
    k_agg<<<gEH, 256, 0, stream>>>(score, z, vbuf, src, dst, qagg, EE);
    k_gemm128<<<gN, 256, 0, stream>>>(qagg, Wo + wo_off, feat, NN, GM_RELU_RES,
                                      nullptr, nullptr, nullptr);
    k_logit<<<1, 256, 0, stream>>>(feat, p1mn, mid, tidx, base_tw, out,
                                   1.0f / NBLK);
  }
}